// GATv2Classifier_73340861547022
// MI455X (gfx1250) — compile-verified
//
#include <hip/hip_runtime.h>
#include <hip/hip_bf16.h>
#include <math.h>

typedef __attribute__((ext_vector_type(2))) float v2f;
typedef __attribute__((ext_vector_type(8))) float v8f;

#define NEG_SLOPE 0.2f

// ---------------------------------------------------------------------------
// fp32 WMMA GEMM: C[M,Ncol] = act(A[M,K] @ B[K,Ncol] + bias)
// grid.x = M/16 row tiles, grid.y = Ncol/64 column chunks
// block = 128 threads (4 waves); each wave owns one 16x16 output tile.
// B chunk (K x 64) staged in LDS. Uses V_WMMA_F32_16X16X4_F32 per ISA layouts:
//   A 16x4:  lanes 0-15 -> M rows, VGPR{0,1} = K{0,1} (lanes<16) / K{2,3} (lanes>=16)
//   C/D:     VGPR v -> M = v (lanes 0-15), M = v+8 (lanes 16-31), N = lane&15
// ---------------------------------------------------------------------------
__global__ void k_wmma_gemm_f32(const float* __restrict__ A,
                                const float* __restrict__ B,
                                const float* __restrict__ bias,
                                float* __restrict__ C,
                                int M, int K, int Ncol, int act) {
  extern __shared__ float sB[]; // K * 64 floats
  const int colbase = blockIdx.y * 64;
  for (int i = threadIdx.x; i < K * 64; i += blockDim.x) {
    int kk = i >> 6, cc = i & 63;
    sB[i] = B[(size_t)kk * Ncol + colbase + cc];
  }
  __syncthreads();

  const int wave = threadIdx.x >> 5;
  const int lane = threadIdx.x & 31;
  const int half = lane >> 4;   // 0: lanes 0-15, 1: lanes 16-31
  const int idx  = lane & 15;
  const int row0 = blockIdx.x * 16;
  const int ccol = colbase + wave * 16 + idx;

  const float* Arow = A + (size_t)(row0 + idx) * K;
  __builtin_prefetch(Arow, 0, 1);              // global_prefetch_b8

  v8f acc = {};
  for (int k = 0; k < K; k += 4) {
    v2f a, b;
    a.x = Arow[k + 2 * half + 0];
    a.y = Arow[k + 2 * half + 1];
    b.x = sB[(k + 2 * half + 0) * 64 + wave * 16 + idx];
    b.y = sB[(k + 2 * half + 1) * 64 + wave * 16 + idx];
    acc = __builtin_amdgcn_wmma_f32_16x16x4_f32(false, a, false, b,
                                                (short)0, acc, false, false);
  }

  union { v8f v; float f[8]; } u;
  u.v = acc;
  const float bv = bias ? bias[ccol] : 0.0f;
#pragma unroll
  for (int v = 0; v < 8; ++v) {
    float o = u.f[v] + bv;
    if (act) o = fmaxf(o, 0.0f);
    C[(size_t)(row0 + 8 * half + v) * Ncol + ccol] = o;
  }
}

// ---------------------------------------------------------------------------
// helpers
// ---------------------------------------------------------------------------
__global__ void k_fill(float* __restrict__ p, float v, long long n) {
  long long t = (long long)blockIdx.x * blockDim.x + threadIdx.x;
  if (t < n) p[t] = v;
}

__device__ __forceinline__ void atomicMaxFloat(float* addr, float value) {
  if (value >= 0.0f)
    atomicMax((int*)addr, __float_as_int(value));
  else
    atomicMin((unsigned int*)addr, __float_as_uint(value));
}

__device__ __forceinline__ float lrelu(float z) {
  return (z > 0.0f) ? z : NEG_SLOPE * z;
}

// ---------------------------------------------------------------------------
// Edge kernels: F = H*C = 128 for both layers. One wave per edge; each lane
// owns one float4 (features 4*lane .. 4*lane+3) -> fully coalesced 512B row
// gathers (global_load_b128). Per-head attention score reduced with a wave32
// xor-shuffle butterfly over the C/4-lane group that owns each head.
// ---------------------------------------------------------------------------

// score[e,h] = sum_c att[h,c] * leaky_relu(xl[src,h,c]+xr[dst,h,c]+ea[e]*we[h,c])
// and atomic segment-max into m[dst,h]
__global__ void k_edge_score(const float4* __restrict__ xl4, const float4* __restrict__ xr4,
                             const float* __restrict__ ea, const float4* __restrict__ we4,
                             const float4* __restrict__ att4,
                             const long long* __restrict__ src, const long long* __restrict__ dst,
                             float* __restrict__ score, float* __restrict__ m,
                             int E_, int H, int Cc) {
  const int wave = threadIdx.x >> 5, lane = threadIdx.x & 31;
  const int e = blockIdx.x * (blockDim.x >> 5) + wave;
  if (e >= E_) return;
  const int s = (int)src[e], d = (int)dst[e];
  const int group = Cc >> 2;                     // lanes per head (4 or 16)
  const float eav = ea[e];

  float4 l = xl4[(size_t)s * 32 + lane];
  float4 r = xr4[(size_t)d * 32 + lane];
  float4 w = we4[lane];
  float4 a = att4[lane];

  float z0 = lrelu(l.x + r.x + eav * w.x);
  float z1 = lrelu(l.y + r.y + eav * w.y);
  float z2 = lrelu(l.z + r.z + eav * w.z);
  float z3 = lrelu(l.w + r.w + eav * w.w);
  float sc = a.x * z0 + a.y * z1 + a.z * z2 + a.w * z3;

  for (int off = 1; off < group; off <<= 1) sc += __shfl_xor(sc, off, 32);

  if ((lane & (group - 1)) == 0) {
    int h = lane / group;
    score[(size_t)e * H + h] = sc;
    atomicMaxFloat(&m[(size_t)d * H + h], sc);
  }
}

// a[e,h] = exp(score - m[dst,h]); atomic segment-sum into denom[dst,h]
__global__ void k_edge_exp(const float* __restrict__ m, const long long* __restrict__ dst,
                           float* __restrict__ score, float* __restrict__ denom,
                           int E_, int H) {
  int t = blockIdx.x * blockDim.x + threadIdx.x;
  if (t >= E_ * H) return;
  int e = t / H, h = t - e * H;
  int d = (int)dst[e];
  float a = __expf(score[t] - m[(size_t)d * H + h]);
  score[t] = a;
  atomicAdd(&denom[(size_t)d * H + h], a);
}

// out[dst,h,c] += xl[src,h,c] * a[e,h] / (denom[dst,h] + eps)
// wave per edge; lane owns features 4*lane..4*lane+3 -> coalesced gather +
// 4 consecutive global_atomic_add_f32 per lane.
__global__ void k_edge_scatter(const float4* __restrict__ xl4, const float* __restrict__ score,
                               const float* __restrict__ denom,
                               const long long* __restrict__ src, const long long* __restrict__ dst,
                               float* __restrict__ out, int E_, int H, int Cc) {
  const int wave = threadIdx.x >> 5, lane = threadIdx.x & 31;
  const int e = blockIdx.x * (blockDim.x >> 5) + wave;
  if (e >= E_) return;
  const int s = (int)src[e], d = (int)dst[e];
  const int group = Cc >> 2;
  const int h = lane / group;
  const float alpha = score[(size_t)e * H + h] /
                      (denom[(size_t)d * H + h] + 1e-16f);
  float4 l = xl4[(size_t)s * 32 + lane];
  float* po = out + (size_t)d * 128 + lane * 4;
  atomicAdd(po + 0, l.x * alpha);
  atomicAdd(po + 1, l.y * alpha);
  atomicAdd(po + 2, l.z * alpha);
  atomicAdd(po + 3, l.w * alpha);
}

// buf[t] = relu(buf[t] + bias[t % F])
__global__ void k_bias_relu(float* __restrict__ buf, const float* __restrict__ bias,
                            long long total, int F) {
  long long t = (long long)blockIdx.x * blockDim.x + threadIdx.x;
  if (t >= total) return;
  buf[t] = fmaxf(buf[t] + bias[(int)(t % F)], 0.0f);
}

// out[n,c] = relu(mean_h acc[n,h,c] + bias[c])
__global__ void k_mean_relu(const float* __restrict__ acc, const float* __restrict__ bias,
                            float* __restrict__ out, int n, int H, int D) {
  int t = blockIdx.x * blockDim.x + threadIdx.x;
  if (t >= n * D) return;
  int node = t / D, c = t - node * D;
  float s = 0.0f;
  for (int h = 0; h < H; ++h) s += acc[(size_t)node * H * D + h * D + c];
  out[t] = fmaxf(s / H + bias[c], 0.0f);
}

// out[n,:] = (resid ? resid[n,:] : 0) + gelu(layernorm(in[n,:]) * g + b)   (D == 64)
// one wave per row, 2 features per lane, wave32 shuffle reduction
__global__ void k_ln_gelu(const float* __restrict__ in, const float* __restrict__ g,
                          const float* __restrict__ b, const float* __restrict__ resid,
                          float* __restrict__ out, int n) {
  const int D = 64;
  int wave = threadIdx.x >> 5, lane = threadIdx.x & 31;
  int node = blockIdx.x * (blockDim.x >> 5) + wave;
  if (node >= n) return;
  const float* row = in + (size_t)node * D;
  float x0 = row[lane], x1 = row[lane + 32];
  float s = x0 + x1;
#pragma unroll
  for (int off = 16; off; off >>= 1) s += __shfl_xor(s, off, 32);
  float mu = s * (1.0f / D);
  float d0 = x0 - mu, d1 = x1 - mu;
  float v = d0 * d0 + d1 * d1;
#pragma unroll
  for (int off = 16; off; off >>= 1) v += __shfl_xor(v, off, 32);
  float rstd = rsqrtf(v * (1.0f / D) + 1e-5f);
  float y0 = d0 * rstd * g[lane] + b[lane];
  float y1 = d1 * rstd * g[lane + 32] + b[lane + 32];
  y0 = 0.5f * y0 * (1.0f + erff(y0 * 0.70710678118654752f));
  y1 = 0.5f * y1 * (1.0f + erff(y1 * 0.70710678118654752f));
  if (resid) {
    y0 += resid[(size_t)node * D + lane];
    y1 += resid[(size_t)node * D + lane + 32];
  }
  out[(size_t)node * D + lane] = y0;
  out[(size_t)node * D + lane + 32] = y1;
}

// ---------------------------------------------------------------------------
static inline int cdiv(long long a, long long b) { return (int)((a + b - 1) / b); }

extern "C" void kernel_launch(void* const* d_in, const int* in_sizes, int n_in,
                              void* d_out, int out_size, void* d_ws, size_t ws_size,
                              hipStream_t stream) {
  const float*     x    = (const float*)d_in[0];
  const long long* ei   = (const long long*)d_in[1];
  const float*     ea   = (const float*)d_in[2];
  const float*     w1l  = (const float*)d_in[3];
  const float*     w1r  = (const float*)d_in[4];
  const float*     w1e  = (const float*)d_in[5];
  const float*     att1 = (const float*)d_in[6];
  const float*     b1   = (const float*)d_in[7];
  const float*     w2l  = (const float*)d_in[8];
  const float*     w2r  = (const float*)d_in[9];
  const float*     w2e  = (const float*)d_in[10];
  const float*     att2 = (const float*)d_in[11];
  const float*     b2   = (const float*)d_in[12];
  const float*     wc   = (const float*)d_in[13];
  const float*     bc   = (const float*)d_in[14];
  const float*     lng  = (const float*)d_in[15];
  const float*     lnb  = (const float*)d_in[16];
  const float*     wr   = (const float*)d_in[17];
  const float*     br   = (const float*)d_in[18];
  const float*     lrg  = (const float*)d_in[19];
  const float*     lrb  = (const float*)d_in[20];

  const int N = in_sizes[0] / 64;          // 50000
  const int E = in_sizes[1] / 2;           // 800000
  const long long* src = ei;
  const long long* dst = ei + E;

  float* ws = (float*)d_ws;
  float* XL = ws;                              // N*128
  float* XR = XL + (size_t)N * 128;            // N*128
  float* SC = XR + (size_t)N * 128;            // E*8
  float* Mb = SC + (size_t)E * 8;              // N*8
  float* Dn = Mb + (size_t)N * 8;              // N*8
  float* AC = Dn + (size_t)N * 8;              // N*128 (acc1 -> h1, then acc2)
  float* H2 = AC + (size_t)N * 128;            // N*64
  float* Tb = H2 + (size_t)N * 64;             // N*64
  float* OUT = (float*)d_out;                  // N*64

  const int BT = 256;
  const int EPB = BT / 32;                     // edges per block (wave per edge)
  const float NEG_INF = -__builtin_huge_valf();

  // ---------------- Layer 1 (H=8, C=16, concat, relu) ----------------
  {
    dim3 g(N / 16, 128 / 64);
    k_wmma_gemm_f32<<<g, 128, 64 * 64 * sizeof(float), stream>>>(x, w1l, nullptr, XL, N, 64, 128, 0);
    k_wmma_gemm_f32<<<g, 128, 64 * 64 * sizeof(float), stream>>>(x, w1r, nullptr, XR, N, 64, 128, 0);
  }
  k_fill<<<cdiv((long long)N * 8, BT), BT, 0, stream>>>(Mb, NEG_INF, (long long)N * 8);
  k_fill<<<cdiv((long long)N * 8, BT), BT, 0, stream>>>(Dn, 0.0f, (long long)N * 8);
  k_fill<<<cdiv((long long)N * 128, BT), BT, 0, stream>>>(AC, 0.0f, (long long)N * 128);
  k_edge_score<<<cdiv(E, EPB), BT, 0, stream>>>((const float4*)XL, (const float4*)XR, ea,
                                                (const float4*)w1e, (const float4*)att1,
                                                src, dst, SC, Mb, E, 8, 16);
  k_edge_exp<<<cdiv((long long)E * 8, BT), BT, 0, stream>>>(Mb, dst, SC, Dn, E, 8);
  k_edge_scatter<<<cdiv(E, EPB), BT, 0, stream>>>((const float4*)XL, SC, Dn, src, dst, AC, E, 8, 16);
  k_bias_relu<<<cdiv((long long)N * 128, BT), BT, 0, stream>>>(AC, b1, (long long)N * 128, 128); // AC = h1

  // ---------------- Layer 2 (H=2, C=64, mean, relu) ----------------
  {
    dim3 g(N / 16, 128 / 64);
    k_wmma_gemm_f32<<<g, 128, 128 * 64 * sizeof(float), stream>>>(AC, w2l, nullptr, XL, N, 128, 128, 0);
    k_wmma_gemm_f32<<<g, 128, 128 * 64 * sizeof(float), stream>>>(AC, w2r, nullptr, XR, N, 128, 128, 0);
  }
  k_fill<<<cdiv((long long)N * 2, BT), BT, 0, stream>>>(Mb, NEG_INF, (long long)N * 2);
  k_fill<<<cdiv((long long)N * 2, BT), BT, 0, stream>>>(Dn, 0.0f, (long long)N * 2);
  k_fill<<<cdiv((long long)N * 128, BT), BT, 0, stream>>>(AC, 0.0f, (long long)N * 128);
  k_edge_score<<<cdiv(E, EPB), BT, 0, stream>>>((const float4*)XL, (const float4*)XR, ea,
                                                (const float4*)w2e, (const float4*)att2,
                                                src, dst, SC, Mb, E, 2, 64);
  k_edge_exp<<<cdiv((long long)E * 2, BT), BT, 0, stream>>>(Mb, dst, SC, Dn, E, 2);
  k_edge_scatter<<<cdiv(E, EPB), BT, 0, stream>>>((const float4*)XL, SC, Dn, src, dst, AC, E, 2, 64);
  k_mean_relu<<<cdiv((long long)N * 64, BT), BT, 0, stream>>>(AC, b2, H2, N, 2, 64); // H2 = h2

  // ---------------- Classifier ----------------
  {
    dim3 g(N / 16, 1);
    k_wmma_gemm_f32<<<g, 128, 64 * 64 * sizeof(float), stream>>>(H2, wc, bc, Tb, N, 64, 64, 0);
    k_ln_gelu<<<cdiv(N, 8), 256, 0, stream>>>(Tb, lng, lnb, nullptr, Tb, N);       // Tb = t
    k_wmma_gemm_f32<<<g, 128, 64 * 64 * sizeof(float), stream>>>(Tb, wr, br, H2, N, 64, 64, 0);
    k_ln_gelu<<<cdiv(N, 8), 256, 0, stream>>>(H2, lrg, lrb, Tb, OUT, N);           // out = t + gelu(ln(v))
  }
}